// BasisDecorrelationLoss_34540126994846
// MI455X (gfx1250) — compile-verified
//
#include <hip/hip_runtime.h>
#include <math.h>

typedef __attribute__((ext_vector_type(2))) float v2f;
typedef __attribute__((ext_vector_type(8))) float v8f;

#define CH        32
#define NPIX      (240 * 320)      /* 76800 */
#define CHUNK     128
#define CHUNKS_PER_WG 10
#define WGS_PER_BATCH 60           /* 60 * 10 * 128 = 76800 */
#define NBATCH    8
#define KPAIRS    (CHUNK / 2)      /* 64 k-pair rows per chunk */
#define PPITCH    33               /* pairs per row (32 + 1 pad) */
#define EPSV      1e-10f

// ---------------------------------------------------------------------------
// Zero the workspace accumulators (ws is poisoned with 0xAA by the harness and
// never re-zeroed between replays, so this must run every launch).
// ---------------------------------------------------------------------------
__global__ void zero_ws_kernel(float* __restrict__ p, int n) {
    int i = blockIdx.x * 256 + threadIdx.x;
    if (i < n) p[i] = 0.0f;
}

// ---------------------------------------------------------------------------
// Main single-pass kernel: per batch, accumulate
//   S_cd = sum_n m[n] * x_c[n] * x_d[n]   (f32 WMMA on Y = X * diag(sqrt m))
//   t_c  = sum_n m[n] * x_c[n],  s_c = sum_n x_c[n],  M = sum_n m[n]
// Double-buffered LDS + software-pipelined global loads + prefetch.
// Chunk loop pinned to unroll 2 (= double-buffer period) to keep VGPRs < 256
// and occupancy high (this kernel is HBM-bandwidth bound).
// ---------------------------------------------------------------------------
__global__ __launch_bounds__(256)
void gram_kernel(const float* __restrict__ x, const float* __restrict__ mask,
                 float* __restrict__ gS, float* __restrict__ gs,
                 float* __restrict__ gt, float* __restrict__ gM)
{
    __shared__ float sm[2][CHUNK];               // sqrt(mask), double buffered
    __shared__ v2f   ylds[2][KPAIRS * PPITCH];   // y k-pairs: [kp][ch] = {y2k,y2k+1}
    __shared__ float red[256];

    const int t    = threadIdx.x;
    const int b    = blockIdx.x / WGS_PER_BATCH;
    const int g    = blockIdx.x % WGS_PER_BATCH;
    const int c    = t >> 3;        // channel 0..31 (fixed per thread)
    const int u    = t & 7;         // sub-slot within channel
    const int lane = t & 31;
    const int wave = t >> 5;

    const float* xb = x    + (size_t)b * CH * NPIX;
    const float* mb = mask + (size_t)b * NPIX;
    const float* xrow = xb + (size_t)c * NPIX;   // this thread's channel row

    v8f acc00 = {}, acc01 = {}, acc10 = {}, acc11 = {};
    float sx = 0.0f, sxm = 0.0f, msum = 0.0f;

    const int chL   = lane & 15;         // operand row|col index
    const int kpoff = (lane >> 4);       // k-pair sub-offset per lane half

    // ---- prologue: loads for chunk 0 ----
    int n0 = g * CHUNKS_PER_WG * CHUNK;
    float4 mreg = make_float4(0.f, 0.f, 0.f, 0.f);
    float4 xm[4];
    if (t < CHUNK / 4) mreg = *(const float4*)(mb + n0 + 4 * t);
    {
        const float* xr = xrow + n0 + u * 16;
#pragma unroll
        for (int q = 0; q < 4; ++q) xm[q] = *(const float4*)(xr + 4 * q);
    }

#pragma unroll 2
    for (int it = 0; it < CHUNKS_PER_WG; ++it) {
        const int p  = it & 1;
        const int n1 = n0 + CHUNK;

        // --- stage sqrt(mask) into buffer p; start next mask load ---
        if (t < CHUNK / 4) {
            msum += mreg.x + mreg.y + mreg.z + mreg.w;
            sm[p][4 * t + 0] = __builtin_amdgcn_sqrtf(mreg.x);
            sm[p][4 * t + 1] = __builtin_amdgcn_sqrtf(mreg.y);
            sm[p][4 * t + 2] = __builtin_amdgcn_sqrtf(mreg.z);
            sm[p][4 * t + 3] = __builtin_amdgcn_sqrtf(mreg.w);
            if (it + 1 < CHUNKS_PER_WG)
                mreg = *(const float4*)(mb + n1 + 4 * t);
        }
        __syncthreads();

        // --- stage y = x * sqrt(m) as interleaved k-pairs (ds_store_b64) ---
#pragma unroll
        for (int q = 0; q < 4; ++q) {
            const int lk = u * 16 + 4 * q;        // even
            const int kp = lk >> 1;
            float s0 = sm[p][lk + 0], s1 = sm[p][lk + 1];
            float s2 = sm[p][lk + 2], s3 = sm[p][lk + 3];
            float y0 = xm[q].x * s0, y1 = xm[q].y * s1;
            float y2 = xm[q].z * s2, y3 = xm[q].w * s3;
            sx  += xm[q].x + xm[q].y + xm[q].z + xm[q].w;
            sxm += y0 * s0 + y1 * s1 + y2 * s2 + y3 * s3;   // = sum x*m
            ylds[p][(kp + 0) * PPITCH + c] = (v2f){ y0, y1 };
            ylds[p][(kp + 1) * PPITCH + c] = (v2f){ y2, y3 };
        }

        // --- software pipeline: global loads for chunk it+1 in flight
        //     across the barrier + WMMA phase; prefetch chunk it+2 ---
        if (it + 1 < CHUNKS_PER_WG) {
            const float* xr = xrow + n1 + u * 16;
#pragma unroll
            for (int q = 0; q < 4; ++q) xm[q] = *(const float4*)(xr + 4 * q);
            if (it + 2 < CHUNKS_PER_WG)
                __builtin_prefetch(xrow + n1 + CHUNK + u * 16, 0, 3);
        }
        __syncthreads();

        // --- WMMA phase on buffer p: wave w owns k-pairs [w*8, w*8+8) ---
        const v2f* yp = &ylds[p][0];
#pragma unroll
        for (int q = 0; q < 4; ++q) {
            const int kp = wave * 8 + q * 2 + kpoff;
            // A 16x4 f32: lane holds M=lane&15, VGPR v -> K = 2*(lane>>4)+v
            // B 4x16 f32: lane holds N=lane&15, VGPR v -> K = 2*(lane>>4)+v
            v2f rl = yp[kp * PPITCH + chL];         // channels 0..15
            v2f rh = yp[kp * PPITCH + chL + 16];    // channels 16..31
            acc00 = __builtin_amdgcn_wmma_f32_16x16x4_f32(
                false, rl, false, rl, (short)0, acc00, false, false);
            acc01 = __builtin_amdgcn_wmma_f32_16x16x4_f32(
                false, rl, false, rh, (short)0, acc01, false, false);
            acc10 = __builtin_amdgcn_wmma_f32_16x16x4_f32(
                false, rh, false, rl, (short)0, acc10, false, false);
            acc11 = __builtin_amdgcn_wmma_f32_16x16x4_f32(
                false, rh, false, rh, (short)0, acc11, false, false);
        }
        n0 = n1;
    }

    // --- flush Gram accumulators: C/D layout: VGPR r, lane L ->
    //     M = r + 8*(L>>4), N = L&15 ---
    {
        float* Sb = gS + b * CH * CH;
        const int Mb = (lane >> 4) * 8;
        const int Nc = lane & 15;
#pragma unroll
        for (int r = 0; r < 8; ++r) {
            atomicAdd(&Sb[(Mb + r)      * CH + Nc     ], acc00[r]);
            atomicAdd(&Sb[(Mb + r)      * CH + Nc + 16], acc01[r]);
            atomicAdd(&Sb[(Mb + r + 16) * CH + Nc     ], acc10[r]);
            atomicAdd(&Sb[(Mb + r + 16) * CH + Nc + 16], acc11[r]);
        }
    }

    // --- per-channel sums (8 threads per channel) ---
    __syncthreads();
    red[t] = sx;
    __syncthreads();
    if (u == 0) {
        float v = 0.0f;
#pragma unroll
        for (int j = 0; j < 8; ++j) v += red[t + j];
        atomicAdd(&gs[b * CH + c], v);
    }
    __syncthreads();
    red[t] = sxm;
    __syncthreads();
    if (u == 0) {
        float v = 0.0f;
#pragma unroll
        for (int j = 0; j < 8; ++j) v += red[t + j];
        atomicAdd(&gt[b * CH + c], v);
    }
    __syncthreads();
    red[t] = (t < 32) ? msum : 0.0f;
    __syncthreads();
    if (t == 0) {
        float v = 0.0f;
#pragma unroll
        for (int j = 0; j < 32; ++j) v += red[j];
        atomicAdd(&gM[b], v);
    }
}

// ---------------------------------------------------------------------------
// Finalize: assemble cov, clamp with EPS, mean of zncc^2 -> scalar.
// One block of 1024 threads; thread t owns pair (c,d) = (t>>5, t&31).
// ---------------------------------------------------------------------------
__global__ __launch_bounds__(1024)
void finalize_kernel(const float* __restrict__ gS, const float* __restrict__ gs,
                     const float* __restrict__ gt, const float* __restrict__ gM,
                     float* __restrict__ out)
{
    __shared__ float red[1024];
    const int t = threadIdx.x;
    const int cc = t >> 5;
    const int d  = t & 31;
    const float invN = 1.0f / (float)NPIX;

    float accum = 0.0f;
    for (int b = 0; b < NBATCH; ++b) {
        const float* Sb = gS + b * CH * CH;
        float M    = gM[b];
        float invM = 1.0f / M;
        float mu_c = gs[b * CH + cc] * invN;
        float mu_d = gs[b * CH + d]  * invN;
        float t_c  = gt[b * CH + cc];
        float t_d  = gt[b * CH + d];

        float cov_cd = (Sb[cc * CH + d]  - mu_c * t_d - mu_d * t_c + mu_c * mu_d * M) * invM;
        float cov_cc = (Sb[cc * CH + cc] - 2.0f * mu_c * t_c + mu_c * mu_c * M) * invM;
        float cov_dd = (Sb[d  * CH + d]  - 2.0f * mu_d * t_d + mu_d * mu_d * M) * invM;
        cov_cd = fmaxf(cov_cd, EPSV);
        cov_cc = fmaxf(cov_cc, EPSV);
        cov_dd = fmaxf(cov_dd, EPSV);
        // zncc^2 = cov_cd^2 / (cov_cc * cov_dd)
        accum += (cov_cd * cov_cd) / (cov_cc * cov_dd);
    }

    red[t] = accum;
    __syncthreads();
    for (int s = 512; s > 0; s >>= 1) {
        if (t < s) red[t] += red[t + s];
        __syncthreads();
    }
    if (t == 0) out[0] = red[0] / (float)(NBATCH * CH * CH);
}

// ---------------------------------------------------------------------------
extern "C" void kernel_launch(void* const* d_in, const int* in_sizes, int n_in,
                              void* d_out, int out_size, void* d_ws, size_t ws_size,
                              hipStream_t stream) {
    const float* x = (const float*)d_in[0];   // depth_basis [8,32,240,320] f32
    const float* m = (const float*)d_in[1];   // mask        [8,1,240,320]  f32
    float* out = (float*)d_out;

    float* ws = (float*)d_ws;
    float* gS = ws;                                    // 8*32*32 = 8192 floats
    float* gs = ws + NBATCH * CH * CH;                 // 256
    float* gt = gs + NBATCH * CH;                      // 256
    float* gM = gt + NBATCH * CH;                      // 8
    const int ztotal = NBATCH * CH * CH + 2 * NBATCH * CH + NBATCH;  // 8712

    zero_ws_kernel<<<(ztotal + 255) / 256, 256, 0, stream>>>(ws, ztotal);
    gram_kernel<<<NBATCH * WGS_PER_BATCH, 256, 0, stream>>>(x, m, gS, gs, gt, gM);
    finalize_kernel<<<1, 1024, 0, stream>>>(gS, gs, gt, gM, out);
}